// Attention_85847806313093
// MI455X (gfx1250) — compile-verified
//
#include <hip/hip_runtime.h>
#include <hip/hip_bf16.h>
#include <stdint.h>

// Problem constants (from reference setup_inputs)
#define Bn 32
#define Sn 4096
#define Hn 1024
#define Rn (Bn * Sn)          // 131072 rows
#define LN_EPS 1e-5f
#define SCHUNK 128            // s-elements per pooling block
#define NCHUNK (Sn / SCHUNK)  // 32

// Workspace layout (float offsets). Total = OFF_ACCUM + Bn*Hn = 428,032 floats ~= 1.72 MB
#define OFF_GW     0                  // gamma*w           [Hn]
#define OFF_SUMGW  1024               // sum(gamma*w)      [1]
#define OFF_DOTBW  1025               // dot(beta, w)      [1]
#define OFF_DENOM  1056               // softmax denom     [Bn]
#define OFF_CSUM   1088               // sum p*rstd*mu     [Bn]
#define OFF_MU     2048               // row means         [Rn]
#define OFF_RSTD   (OFF_MU + Rn)      // row rstd          [Rn]
#define OFF_ACOEF  (OFF_MU + 2 * Rn)  // scores -> p*rstd  [Rn]
#define OFF_ACCUM  (OFF_MU + 3 * Rn)  // GEMV accumulators [Bn*Hn]

typedef unsigned int u32x4 __attribute__((ext_vector_type(4)));
typedef int          i32x8 __attribute__((ext_vector_type(8)));
typedef int          i32x4 __attribute__((ext_vector_type(4)));

// ---------------- wave32 / block reductions ----------------
__device__ __forceinline__ float wave_sum(float v) {
#pragma unroll
  for (int m = 16; m > 0; m >>= 1) v += __shfl_xor(v, m, 32);
  return v;
}
__device__ __forceinline__ float wave_max(float v) {
#pragma unroll
  for (int m = 16; m > 0; m >>= 1) v = fmaxf(v, __shfl_xor(v, m, 32));
  return v;
}
__device__ __forceinline__ float block_sum(float v, float* sh) {
  const int lane = threadIdx.x & 31, wv = threadIdx.x >> 5;
  v = wave_sum(v);
  if (lane == 0) sh[wv] = v;
  __syncthreads();
  if (wv == 0) {
    float t = (lane < 8) ? sh[lane] : 0.0f;
    t = wave_sum(t);
    if (lane == 0) sh[0] = t;
  }
  __syncthreads();
  float r = sh[0];
  __syncthreads();
  return r;
}
__device__ __forceinline__ float block_max(float v, float* sh) {
  const int lane = threadIdx.x & 31, wv = threadIdx.x >> 5;
  v = wave_max(v);
  if (lane == 0) sh[wv] = v;
  __syncthreads();
  if (wv == 0) {
    float t = (lane < 8) ? sh[lane] : -3.4e38f;
    t = wave_max(t);
    if (lane == 0) sh[0] = t;
  }
  __syncthreads();
  float r = sh[0];
  __syncthreads();
  return r;
}

// ---------------- k0: fold gamma into w; scalar terms ----------------
__global__ void __launch_bounds__(256) k0_prep(const float* __restrict__ gamma,
                                               const float* __restrict__ beta,
                                               const float* __restrict__ w,
                                               float* __restrict__ wsf) {
  __shared__ float sh[8];
  const int t = threadIdx.x;  // 256 threads * float4 = 1024
  float4 g  = ((const float4*)gamma)[t];
  float4 b4 = ((const float4*)beta)[t];
  float4 wv = ((const float4*)w)[t];
  float4 gw;
  gw.x = g.x * wv.x; gw.y = g.y * wv.y; gw.z = g.z * wv.z; gw.w = g.w * wv.w;
  ((float4*)(wsf + OFF_GW))[t] = gw;
  float sg = gw.x + gw.y + gw.z + gw.w;
  float db = b4.x * wv.x + b4.y * wv.y + b4.z * wv.z + b4.w * wv.w;
  sg = block_sum(sg, sh);
  db = block_sum(db, sh);
  if (t == 0) { wsf[OFF_SUMGW] = sg; wsf[OFF_DOTBW] = db; }
}

// ---------------- k1: fused LN-stats + score (first full read of x) ----------------
// One wave32 per 4KB row; 8 rows per 256-thread block.
__global__ void __launch_bounds__(256) k1_stats(const float* __restrict__ x,
                                                float* __restrict__ wsf) {
  const int lane = threadIdx.x & 31;
  const int wv   = threadIdx.x >> 5;
  const int r    = blockIdx.x * 8 + wv;  // row index in [0, Rn)
  const float4* row4 = (const float4*)(x + (size_t)r * Hn);
  const float4* gw4  = (const float4*)(wsf + OFF_GW);
  float s = 0.f, ss = 0.f, d = 0.f;
#pragma unroll
  for (int i = 0; i < 8; ++i) {
    const int idx = i * 32 + lane;       // 512B contiguous per wave per step
    float4 v = row4[idx];
    float4 g = gw4[idx];
    s  += v.x + v.y + v.z + v.w;
    ss += v.x * v.x + v.y * v.y + v.z * v.z + v.w * v.w;
    d  += v.x * g.x + v.y * g.y + v.z * g.z + v.w * g.w;
  }
  s = wave_sum(s); ss = wave_sum(ss); d = wave_sum(d);
  if (lane == 0) {
    const float invH = 1.0f / (float)Hn;
    float mu   = s * invH;
    float var  = fmaxf(ss * invH - mu * mu, 0.0f);
    float rstd = rsqrtf(var + LN_EPS);
    wsf[OFF_MU + r]    = mu;
    wsf[OFF_RSTD + r]  = rstd;
    // score = rstd*(dot(x,gw) - mu*sum(gw)) + dot(beta,w)
    wsf[OFF_ACOEF + r] = rstd * (d - mu * wsf[OFF_SUMGW]) + wsf[OFF_DOTBW];
  }
}

// ---------------- k2: per-batch max/clamp/softmax; emit acoef = p*rstd ----------------
__global__ void __launch_bounds__(256) k2_softmax(float* __restrict__ wsf) {
  __shared__ float sh[8];
  const int b = blockIdx.x, t = threadIdx.x;
  // Re-zero the pooling accumulators every call (graph-replay safe).
  float4 z = {0.f, 0.f, 0.f, 0.f};
  ((float4*)(wsf + OFF_ACCUM + (size_t)b * Hn))[t] = z;

  float*       sc_arr = wsf + OFF_ACOEF + (size_t)b * Sn;  // in-place score->acoef
  const float* rs_arr = wsf + OFF_RSTD  + (size_t)b * Sn;
  const float* mu_arr = wsf + OFF_MU    + (size_t)b * Sn;

  float sc[16];
  float m = -3.4e38f;
#pragma unroll
  for (int i = 0; i < 16; ++i) {
    sc[i] = sc_arr[t + i * 256];
    m = fmaxf(m, sc[i]);
  }
  m = block_max(m, sh);
  float pd = 0.f, pc = 0.f;
#pragma unroll
  for (int i = 0; i < 16; ++i) {
    const int idx = t + i * 256;
    float e  = __expf(fminf(fmaxf(sc[i] - m, -10.f), 10.f));
    float rs = rs_arr[idx];
    sc_arr[idx] = e * rs;          // unnormalized pooling coefficient
    pd += e;                       // softmax denominator
    pc += e * rs * mu_arr[idx];    // sum p*rstd*mu (mean-removal term)
  }
  pd = block_sum(pd, sh);
  pc = block_sum(pc, sh);
  if (t == 0) { wsf[OFF_DENOM + b] = pd; wsf[OFF_CSUM + b] = pc; }
}

// ---------------- k3: weighted-sum pooling (second full read of x) ----------------
// grid = (NCHUNK, Bn); TDM stages the 128-float coefficient tile into LDS.
__global__ void __launch_bounds__(256) k3_pool(const float* __restrict__ x,
                                               float* __restrict__ wsf) {
  __shared__ float aS[SCHUNK];
  const int chunk = blockIdx.x, b = blockIdx.y, t = threadIdx.x;
  const float* asrc = wsf + OFF_ACOEF + (size_t)b * Sn + (size_t)chunk * SCHUNK;

#if defined(__gfx1250__)
  if (t < 32) {  // wave 0 issues one TDM descriptor: 1-D tile, 128 x f32, global->LDS
    unsigned long long ga = (unsigned long long)(uintptr_t)asrc;
    unsigned lds = (unsigned)(uintptr_t)(void*)aS;
    u32x4 g0;
    g0[0] = 1u;                                               // count=1, user descriptor
    g0[1] = lds;                                              // lds_addr (bytes)
    g0[2] = (unsigned)(ga & 0xFFFFFFFFu);                     // global_addr[31:0]
    g0[3] = (unsigned)((ga >> 32) & 0x1FFFFFFull) | (2u << 30);  // addr[56:32] | type=2
    i32x8 g1;
    g1[0] = (int)(2u << 16);                 // data_size = 4B
    g1[1] = (int)((unsigned)SCHUNK << 16);   // tensor_dim0 = 128 (bits 79:48 low half)
    g1[2] = (int)(1u << 16);                 // tensor_dim0 hi = 0 ; tensor_dim1 = 1
    g1[3] = (int)((unsigned)SCHUNK << 16);   // tile_dim0 = 128
    g1[4] = 0;                               // tile_dim1 = tile_dim2 = 0 (unused)
    g1[5] = SCHUNK;                          // tensor_dim0_stride = 128
    g1[6] = 0;
    g1[7] = 0;
    i32x4 gz4 = {0, 0, 0, 0};                // groups 2/3 unused (<=2D tensor)
    i32x8 gz8 = {0, 0, 0, 0, 0, 0, 0, 0};    // extra group (6-arg toolchain form)
    __builtin_amdgcn_tensor_load_to_lds(g0, g1, gz4, gz4, gz8, 0);
    __builtin_amdgcn_s_wait_tensorcnt(0);
  }
#else
  if (t < SCHUNK) aS[t] = asrc[t];
#endif
  __syncthreads();

  // 256 threads cover all 1024 h as float4; each s-iteration reads one 4KB row coalesced.
  const float4* xp = (const float4*)x + ((size_t)b * Sn + (size_t)chunk * SCHUNK) * (Hn / 4);
  float4 acc = {0.f, 0.f, 0.f, 0.f};
#pragma unroll 4
  for (int s = 0; s < SCHUNK; ++s) {
    const float a = aS[s];                   // LDS broadcast
    float4 v = xp[(size_t)s * (Hn / 4) + t];
    acc.x = fmaf(a, v.x, acc.x);
    acc.y = fmaf(a, v.y, acc.y);
    acc.z = fmaf(a, v.z, acc.z);
    acc.w = fmaf(a, v.w, acc.w);
  }
  float* accum = wsf + OFF_ACCUM + (size_t)b * Hn + (size_t)t * 4;
  unsafeAtomicAdd(accum + 0, acc.x);
  unsafeAtomicAdd(accum + 1, acc.y);
  unsafeAtomicAdd(accum + 2, acc.z);
  unsafeAtomicAdd(accum + 3, acc.w);
}

// ---------------- k4: finalize out = gamma*(S - csum)/denom + beta ----------------
__global__ void __launch_bounds__(256) k4_final(const float* __restrict__ gamma,
                                                const float* __restrict__ beta,
                                                const float* __restrict__ wsf,
                                                float* __restrict__ out) {
  const int b = blockIdx.x, t = threadIdx.x;
  const float inv = 1.0f / wsf[OFF_DENOM + b];
  const float cs  = wsf[OFF_CSUM + b];
  float4 g  = ((const float4*)gamma)[t];
  float4 bt = ((const float4*)beta)[t];
  float4 a  = ((const float4*)(wsf + OFF_ACCUM + (size_t)b * Hn))[t];
  float4 o;
  o.x = fmaf(g.x, (a.x - cs) * inv, bt.x);
  o.y = fmaf(g.y, (a.y - cs) * inv, bt.y);
  o.z = fmaf(g.z, (a.z - cs) * inv, bt.z);
  o.w = fmaf(g.w, (a.w - cs) * inv, bt.w);
  ((float4*)out)[b * 256 + t] = o;
}

extern "C" void kernel_launch(void* const* d_in, const int* in_sizes, int n_in,
                              void* d_out, int out_size, void* d_ws, size_t ws_size,
                              hipStream_t stream) {
  const float* x     = (const float*)d_in[0];  // [32,4096,1024] f32
  const float* gamma = (const float*)d_in[1];  // [1024]
  const float* beta  = (const float*)d_in[2];  // [1024]
  const float* w     = (const float*)d_in[3];  // [1024]
  float* out = (float*)d_out;                  // [32,1024]
  float* wsf = (float*)d_ws;                   // needs (OFF_ACCUM + Bn*Hn)*4 ~= 1.72 MB

  k0_prep<<<1, 256, 0, stream>>>(gamma, beta, w, wsf);
  k1_stats<<<Rn / 8, 256, 0, stream>>>(x, wsf);
  k2_softmax<<<Bn, 256, 0, stream>>>(wsf);
  dim3 g3(NCHUNK, Bn);
  k3_pool<<<g3, 256, 0, stream>>>(x, wsf);
  k4_final<<<Bn, 256, 0, stream>>>(gamma, beta, wsf, out);
}